// GATLayer_29257317220551
// MI455X (gfx1250) — compile-verified
//
#include <hip/hip_runtime.h>
#include <hip/hip_bf16.h>

typedef __attribute__((ext_vector_type(2))) float v2f;
typedef __attribute__((ext_vector_type(8))) float v8f;

#define NN   50000
#define EE   800000
#define DIN  128
#define DOUT 64

// ---------------------------------------------------------------------------
// Kernel 1: h = X @ W + b_W  via V_WMMA_F32_16X16X4_F32 (exact fp32 math).
// One wave computes a 16x64 tile of h. 4 accumulators (16x16 each),
// K-loop 128/4 = 32 steps -> 128 WMMAs per wave.
// ---------------------------------------------------------------------------
__global__ __launch_bounds__(256) void gat_gemm_wmma(
    const float* __restrict__ X,   // [NN, DIN]
    const float* __restrict__ W,   // [DIN, DOUT]
    const float* __restrict__ bW,  // [DOUT]
    float* __restrict__ H)         // [NN, DOUT]
{
    const int lane  = threadIdx.x & 31;
    const int wave  = blockIdx.x * (blockDim.x >> 5) + (threadIdx.x >> 5);
    const int row0  = wave * 16;
    if (row0 >= NN) return;   // wave-uniform: EXEC stays all-ones for WMMA

    const int m     = lane & 15;   // A-row / B-col / D-col within tile
    const int khalf = lane >> 4;   // K split: lanes 0-15 -> K{0,1}, 16-31 -> K{2,3}

    const float* xrow = X + (size_t)(row0 + m) * DIN + khalf * 2;

    v8f acc0 = {}; v8f acc1 = {}; v8f acc2 = {}; v8f acc3 = {};

    for (int k0 = 0; k0 < DIN; k0 += 4) {
        // A fragment (16x4): a.x = X[m][k0+2*khalf], a.y = X[m][k0+2*khalf+1]
        v2f a;
        a.x = xrow[k0 + 0];
        a.y = xrow[k0 + 1];

        const int kb = k0 + khalf * 2;
        const float* wr0 = W + (size_t)kb * DOUT;        // row kb
        const float* wr1 = wr0 + DOUT;                   // row kb+1

        // 4 N-tiles of B (4x16 each): b.x = W[kb][n], b.y = W[kb+1][n]
        v2f b0; b0.x = wr0[ 0 + m]; b0.y = wr1[ 0 + m];
        v2f b1; b1.x = wr0[16 + m]; b1.y = wr1[16 + m];
        v2f b2; b2.x = wr0[32 + m]; b2.y = wr1[32 + m];
        v2f b3; b3.x = wr0[48 + m]; b3.y = wr1[48 + m];

        acc0 = __builtin_amdgcn_wmma_f32_16x16x4_f32(false, a, false, b0, (short)0, acc0, false, false);
        acc1 = __builtin_amdgcn_wmma_f32_16x16x4_f32(false, a, false, b1, (short)0, acc1, false, false);
        acc2 = __builtin_amdgcn_wmma_f32_16x16x4_f32(false, a, false, b2, (short)0, acc2, false, false);
        acc3 = __builtin_amdgcn_wmma_f32_16x16x4_f32(false, a, false, b3, (short)0, acc3, false, false);
    }

    // D layout: VGPR r -> row (r + 8*khalf), col = lane&15 within each 16-wide tile.
    #pragma unroll
    for (int r = 0; r < 8; ++r) {
        const int mm = r + 8 * khalf;
        float* orow = H + (size_t)(row0 + mm) * DOUT;
        orow[ 0 + m] = acc0[r] + bW[ 0 + m];
        orow[16 + m] = acc1[r] + bW[16 + m];
        orow[32 + m] = acc2[r] + bW[32 + m];
        orow[48 + m] = acc3[r] + bW[48 + m];
    }
}

// ---------------------------------------------------------------------------
// Kernel 2: per-node attention scores s1 = h.a1 + a1b, s2 = h.a2 + a2b.
// ---------------------------------------------------------------------------
__global__ __launch_bounds__(256) void gat_scores(
    const float* __restrict__ H,
    const float* __restrict__ a1w, const float* __restrict__ a1b,
    const float* __restrict__ a2w, const float* __restrict__ a2b,
    float* __restrict__ s1, float* __restrict__ s2)
{
    const int i = blockIdx.x * blockDim.x + threadIdx.x;
    if (i >= NN) return;
    const float* hr = H + (size_t)i * DOUT;
    float d1 = 0.f, d2 = 0.f;
    #pragma unroll 8
    for (int k = 0; k < DOUT; ++k) {
        const float hv = hr[k];
        d1 = fmaf(hv, a1w[k], d1);
        d2 = fmaf(hv, a2w[k], d2);
    }
    s1[i] = d1 + a1b[0];
    s2[i] = d2 + a2b[0];
}

// ---------------------------------------------------------------------------
// Kernel 3: one wave32 per node. row[] is sorted -> binary search the edge
// range, then: pass1 lane-parallel max, pass2 lane-parallel sum(exp), both
// reduced with wave32 shfl_xor; pass3 feature-parallel accumulation
// (lane owns features {lane, lane+32}). No atomics, fully deterministic.
// ---------------------------------------------------------------------------
__global__ __launch_bounds__(256) void gat_aggregate(
    const float* __restrict__ H,
    const float* __restrict__ s1, const float* __restrict__ s2,
    const int* __restrict__ row, const int* __restrict__ col,
    float* __restrict__ out)
{
    const int lane = threadIdx.x & 31;
    const int node = blockIdx.x * (blockDim.x >> 5) + (threadIdx.x >> 5);
    if (node >= NN) return;

    // lower_bound(row, node) and lower_bound(row, node+1)
    int lo = 0, hi = EE;
    while (lo < hi) { const int mid = (lo + hi) >> 1; if (row[mid] < node) lo = mid + 1; else hi = mid; }
    const int start = lo;
    hi = EE;
    while (lo < hi) { const int mid = (lo + hi) >> 1; if (row[mid] < node + 1) lo = mid + 1; else hi = mid; }
    const int end = lo;

    float o0 = 0.f, o1 = 0.f;
    if (end > start) {
        const float si = s1[node];

        // pass 1: row max of leaky_relu logits
        float mloc = -__builtin_inff();
        for (int e = start + lane; e < end; e += 32) {
            const float x  = si + s2[col[e]];
            const float lg = x > 0.f ? x : 0.01f * x;
            mloc = fmaxf(mloc, lg);
        }
        #pragma unroll
        for (int off = 16; off > 0; off >>= 1)
            mloc = fmaxf(mloc, __shfl_xor(mloc, off, 32));

        // pass 2: denom = sum exp(logit - max)
        float dloc = 0.f;
        for (int e = start + lane; e < end; e += 32) {
            const float x  = si + s2[col[e]];
            const float lg = x > 0.f ? x : 0.01f * x;
            dloc += __expf(lg - mloc);
        }
        #pragma unroll
        for (int off = 16; off > 0; off >>= 1)
            dloc += __shfl_xor(dloc, off, 32);

        const float inv_denom = 1.f / dloc;

        // pass 3: out[node] = sum alpha * h[col]  (alpha uniform across lanes;
        // h gathers hit L2 since h is only 12.8 MB)
        for (int e = start; e < end; ++e) {
            const int j    = col[e];
            const float x  = si + s2[j];
            const float lg = x > 0.f ? x : 0.01f * x;
            const float alpha = __expf(lg - mloc) * inv_denom;
            const float* hj = H + (size_t)j * DOUT;
            o0 = fmaf(alpha, hj[lane],      o0);
            o1 = fmaf(alpha, hj[lane + 32], o1);
        }
    }
    out[(size_t)node * DOUT + lane]      = o0;
    out[(size_t)node * DOUT + lane + 32] = o1;
}

// ---------------------------------------------------------------------------
extern "C" void kernel_launch(void* const* d_in, const int* in_sizes, int n_in,
                              void* d_out, int out_size, void* d_ws, size_t ws_size,
                              hipStream_t stream) {
    const float* X   = (const float*)d_in[0];  // features [NN, DIN]
    const float* W   = (const float*)d_in[1];  // [DIN, DOUT]
    const float* bW  = (const float*)d_in[2];  // [DOUT]
    const float* a1w = (const float*)d_in[3];  // [DOUT]
    const float* a1b = (const float*)d_in[4];  // [1]
    const float* a2w = (const float*)d_in[5];  // [DOUT]
    const float* a2b = (const float*)d_in[6];  // [1]
    const int*   row = (const int*)d_in[7];    // [EE] sorted
    const int*   col = (const int*)d_in[8];    // [EE]
    float* out = (float*)d_out;                // [NN, DOUT]

    // workspace layout: H (NN*DOUT) | s1 (NN) | s2 (NN)
    float* H  = (float*)d_ws;
    float* s1 = H + (size_t)NN * DOUT;
    float* s2 = s1 + NN;

    // GEMM: 3125 row-tiles, 8 waves (=tiles) per 256-thread block
    const int tiles = (NN + 15) / 16;                    // 3125
    gat_gemm_wmma<<<(tiles + 7) / 8, 256, 0, stream>>>(X, W, bW, H);

    gat_scores<<<(NN + 255) / 256, 256, 0, stream>>>(H, a1w, a1b, a2w, a2b, s1, s2);

    // 1 wave per node, 8 waves per block
    gat_aggregate<<<(NN + 7) / 8, 256, 0, stream>>>(H, s1, s2, row, col, out);
}